// LoRAWrapperBase_24378234372410
// MI455X (gfx1250) — compile-verified
//
#include <hip/hip_runtime.h>
#include <hip/hip_bf16.h>
#include <stdint.h>

typedef __attribute__((ext_vector_type(2))) float f32x2;
typedef __attribute__((ext_vector_type(4))) float f32x4;
typedef __attribute__((ext_vector_type(8))) float f32x8;

#define D_IN   4096
#define D_OUT  4096
#define S_LEN  2048
#define RANK   16
#define SCALE  2.0f

#define KT     32
#define XPAD   36   // padded LDS row stride (words): 36n mod 64 distinct (conflict-free), 144B rows (16B aligned)
#define RPAD   20   // padded LDS row stride (words): 20n mod 64 distinct, 80B rows (16B aligned)

// Async global -> LDS copy, 16 bytes per active lane. Tracked by ASYNCcnt.
// LDS destination VGPR holds the byte offset within the LDS allocation, which
// equals the low 32 bits of the generic (flat) shared-memory address.
__device__ __forceinline__ void async_copy_b128(uint32_t lds_off, const void* gptr) {
    uint64_t ga = (uint64_t)(uintptr_t)gptr;
    asm volatile("global_load_async_to_lds_b128 %0, %1, off"
                 :: "v"(lds_off), "v"(ga) : "memory");
}

__device__ __forceinline__ void wait_async_zero() {
    asm volatile("s_wait_asynccnt 0" ::: "memory");
}

__device__ __forceinline__ uint32_t lds_off32(const void* p) {
    return (uint32_t)(uintptr_t)p;
}

__global__ __launch_bounds__(256)
void lora_fused_gemm_kernel(const float* __restrict__ x,
                            const int* __restrict__ expert_ids,
                            const float* __restrict__ W,
                            const float* __restrict__ bias,
                            const float* __restrict__ lora_a,
                            const float* __restrict__ lora_b,
                            float* __restrict__ out)
{
    __shared__ float xs[2][128 * XPAD];  // x tile:      128 rows x 32 k (double buffered)
    __shared__ float ws[2][128 * XPAD];  // W tile:      128 cols x 32 k (double buffered)
    __shared__ float las[2][16 * XPAD];  // lora_a tile:  16 r   x 32 k (double buffered)
    __shared__ float ils[128 * RPAD];    // scaled inter: 128 rows x 16 r
    __shared__ float lbs[128 * RPAD];    // lora_b tile:  128 cols x 16 r

    const int tid  = threadIdx.x;
    const int lane = tid & 31;
    const int wave = tid >> 5;
    const int wm   = wave & 3;        // 4 wave-rows of 32 output rows
    const int wn   = wave >> 2;       // 2 wave-cols of 64 output cols
    const int hi   = (lane >> 4) & 1; // half-wave selects K pair inside fragments
    const int l16  = lane & 15;

    const int nbase   = blockIdx.x * 128;
    const int rowbase = blockIdx.y * 128;
    const int batch   = rowbase / S_LEN;      // 128 | 2048 -> tiles never straddle a batch
    const int expert  = expert_ids[batch];
    const float* la = lora_a + (size_t)expert * RANK * D_IN;
    const float* lb = lora_b + (size_t)expert * D_OUT * RANK;

    f32x8 acc[2][4];
    f32x8 inter[2];
    #pragma unroll
    for (int sm = 0; sm < 2; ++sm) {
        inter[sm] = {};
        #pragma unroll
        for (int sn = 0; sn < 4; ++sn) acc[sm][sn] = {};
    }

    // staging indices: 256 threads x 4 rows x one b128 each cover a 128x32 f32 tile
    const int lr = tid >> 3;        // 0..31
    const int lc = (tid & 7) * 4;   // 0,4,...,28

    auto stage = [&](int k0, int b) {
        #pragma unroll
        for (int rr = 0; rr < 4; ++rr) {
            const int row = lr + rr * 32;
            async_copy_b128(lds_off32(&xs[b][row * XPAD + lc]),
                            x + (size_t)(rowbase + row) * D_IN + k0 + lc);
            async_copy_b128(lds_off32(&ws[b][row * XPAD + lc]),
                            W + (size_t)(nbase + row) * D_IN + k0 + lc);
        }
        if (tid < 128) {
            const int r = tid >> 3;  // 0..15
            async_copy_b128(lds_off32(&las[b][r * XPAD + lc]),
                            la + (size_t)r * D_IN + k0 + lc);
        }
    };

    stage(0, 0);  // prologue: fill buffer 0

    for (int k0 = 0; k0 < D_IN; k0 += KT) {
        const int buf = (k0 / KT) & 1;
        wait_async_zero();    // our loads for `buf` are complete
        __syncthreads();      // everyone's loads done; prev compute done -> other buffer reusable
        if (k0 + KT < D_IN) stage(k0 + KT, buf ^ 1);

        #pragma unroll
        for (int ks = 0; ks < KT; ks += 4) {
            f32x2 afrag[2], bfrag[4];
            #pragma unroll
            for (int sm = 0; sm < 2; ++sm)
                afrag[sm] = *(const f32x2*)&xs[buf][(wm * 32 + sm * 16 + l16) * XPAD + ks + 2 * hi];
            #pragma unroll
            for (int sn = 0; sn < 4; ++sn)
                bfrag[sn] = *(const f32x2*)&ws[buf][(wn * 64 + sn * 16 + l16) * XPAD + ks + 2 * hi];
            #pragma unroll
            for (int sm = 0; sm < 2; ++sm)
                #pragma unroll
                for (int sn = 0; sn < 4; ++sn)
                    acc[sm][sn] = __builtin_amdgcn_wmma_f32_16x16x4_f32(
                        false, afrag[sm], false, bfrag[sn], (short)0, acc[sm][sn], false, false);
            if (wn == 0) {  // wave-uniform branch; EXEC all-1s inside
                f32x2 lafrag = *(const f32x2*)&las[buf][l16 * XPAD + ks + 2 * hi];
                #pragma unroll
                for (int sm = 0; sm < 2; ++sm)
                    inter[sm] = __builtin_amdgcn_wmma_f32_16x16x4_f32(
                        false, afrag[sm], false, lafrag, (short)0, inter[sm], false, false);
            }
        }
    }

    // ---- LoRA epilogue: delta = (SCALE*inter) @ lb^T, accumulated via WMMA ----
    {   // async-stage lora_b tile: 128 cols x 16 r
        const int n = tid >> 1;
        const int half = tid & 1;
        #pragma unroll
        for (int i = 0; i < 2; ++i)
            async_copy_b128(lds_off32(&lbs[n * RPAD + half * 8 + i * 4]),
                            lb + (size_t)(nbase + n) * RANK + half * 8 + i * 4);
    }
    if (wn == 0) {
        // C-layout: VGPR e holds M=e (lanes 0-15) / M=e+8 (lanes 16-31); N index = l16 = r
        #pragma unroll
        for (int sm = 0; sm < 2; ++sm)
            #pragma unroll
            for (int e = 0; e < 8; ++e)
                ils[(wm * 32 + sm * 16 + e + 8 * hi) * RPAD + l16] = inter[sm][e] * SCALE;
    }
    wait_async_zero();
    __syncthreads();

    #pragma unroll
    for (int kk = 0; kk < RANK; kk += 4) {
        f32x2 afrag[2], bfrag[4];
        #pragma unroll
        for (int sm = 0; sm < 2; ++sm)
            afrag[sm] = *(const f32x2*)&ils[(wm * 32 + sm * 16 + l16) * RPAD + kk + 2 * hi];
        #pragma unroll
        for (int sn = 0; sn < 4; ++sn)
            bfrag[sn] = *(const f32x2*)&lbs[(wn * 64 + sn * 16 + l16) * RPAD + kk + 2 * hi];
        #pragma unroll
        for (int sm = 0; sm < 2; ++sm)
            #pragma unroll
            for (int sn = 0; sn < 4; ++sn)
                acc[sm][sn] = __builtin_amdgcn_wmma_f32_16x16x4_f32(
                    false, afrag[sm], false, bfrag[sn], (short)0, acc[sm][sn], false, false);
    }

    // ---- bias + store ----
    #pragma unroll
    for (int sn = 0; sn < 4; ++sn) {
        const int col = nbase + wn * 64 + sn * 16 + l16;
        const float bb = bias[col];
        #pragma unroll
        for (int sm = 0; sm < 2; ++sm) {
            const int rb = rowbase + wm * 32 + sm * 16 + 8 * hi;
            #pragma unroll
            for (int e = 0; e < 8; ++e)
                out[(size_t)(rb + e) * D_OUT + col] = acc[sm][sn][e] + bb;
        }
    }
}

extern "C" void kernel_launch(void* const* d_in, const int* in_sizes, int n_in,
                              void* d_out, int out_size, void* d_ws, size_t ws_size,
                              hipStream_t stream) {
    const float* x        = (const float*)d_in[0];
    const int*   expert   = (const int*)d_in[1];
    const float* W        = (const float*)d_in[2];
    const float* b        = (const float*)d_in[3];
    const float* lora_a   = (const float*)d_in[4];
    const float* lora_b   = (const float*)d_in[5];
    float* out = (float*)d_out;

    dim3 grid(D_OUT / 128, (8 * S_LEN) / 128);  // (32, 128)
    dim3 block(256);
    lora_fused_gemm_kernel<<<grid, block, 0, stream>>>(x, expert, W, b, lora_a, lora_b, out);
}